// DepthLSTM_87729001988480
// MI455X (gfx1250) — compile-verified
//
#include <hip/hip_runtime.h>

// Problem dims (fixed by the reference).
#define T_LEN   4096
#define C_CH    256
#define B_SZ    32
#define TT      64                    // timesteps per LDS tile
#define ROWS    32                    // sequences per block == one wave32
#define ROWDW   65                    // padded LDS row stride in dwords (64 data + 1 pad)
#define BUFDW   (ROWS * ROWDW)        // 2080 dwords per tile buffer
#define BUFBYTES (BUFDW * 4)          // 8320 bytes
#define NTILE   (T_LEN / TT)          // 64 tiles per sequence

typedef unsigned int u32x4 __attribute__((ext_vector_type(4)));
typedef int          i32x8 __attribute__((ext_vector_type(8)));
typedef int          i32x4 __attribute__((ext_vector_type(4)));
typedef float        f32x4 __attribute__((ext_vector_type(4)));

#if defined(__HIP_DEVICE_COMPILE__) && __has_builtin(__builtin_amdgcn_tensor_load_to_lds)
#define HAVE_TDM 1
#else
#define HAVE_TDM 0
#endif

// ---------------------------------------------------------------------------
// Fast tanh: gfx1250 has V_TANH_F32 (transcendental unit). Sigmoid is folded
// into tanh at the call sites: sigma(x) = 0.5*tanh(0.5x) + 0.5 with the 0.5
// pre-multiplied into the i/f/o gate weights.
// ---------------------------------------------------------------------------
__device__ __forceinline__ float fast_tanh(float x) {
#if defined(__HIP_DEVICE_COMPILE__)
#if __has_builtin(__builtin_amdgcn_tanhf)
  return __builtin_amdgcn_tanhf(x);
#else
  // tanh(x) = sign(x) * (1 - 2/(exp2(2*log2e*|x|)+1)); saturates correctly.
  float ax = __builtin_fabsf(x);
  float a  = __builtin_amdgcn_exp2f(ax * 2.8853900817779268f); // 2*log2(e)
  float r  = 1.0f - 2.0f * __builtin_amdgcn_rcpf(a + 1.0f);
  return __builtin_copysignf(r, x);
#endif
#else
  return x; // host pass never executes device math
#endif
}

#if HAVE_TDM
// ---------------------------------------------------------------------------
// TDM 2D tile load: rows (sequences) x TT timesteps, global row stride = T.
// LDS padding: 1 dword after every 64 dwords -> 65-dword row stride, which
// makes per-timestep lane reads (addr = lane*65 + t) bank-conflict free.
// Descriptor layout per CDNA5 ISA ch.8 (group0: 4 dw, group1: 8 dw).
// This toolchain's builtin takes 6 args (g0, g1, g2, g3, extra, cpol);
// groups 2/3 and the trailing vector are unused for a 2D tile -> zeros.
// ---------------------------------------------------------------------------
__device__ __forceinline__ void tdm_load_tile(unsigned lds_byte_off,
                                              const float* gptr) {
  unsigned long long ga = (unsigned long long)(__UINTPTR_TYPE__)gptr;
  u32x4 g0;
  g0.x = 1u;                                        // count=1, user desc, no gather
  g0.y = lds_byte_off;                              // lds_addr (bytes)
  g0.z = (unsigned)(ga & 0xffffffffull);            // global_addr[31:0]
  g0.w = ((unsigned)(ga >> 32) & 0x1ffffffu)        // global_addr[56:32]
         | (2u << 30);                              // type = 2 ("image")
  i32x8 g1;
  g1[0] = (int)((2u << 16)                          // data_size = 4B
              | (1u << 20)                          // pad_enable
              | (5u << 22)                          // pad_interval: 64 dwords
              | (0u << 25));                        // pad_amount: 1 dword
  g1[1] = (int)((T_LEN & 0xffffu) << 16);           // tensor_dim0 lo16 (=4096)
  g1[2] = (int)((T_LEN >> 16) | ((ROWS & 0xffffu) << 16)); // td0 hi | tensor_dim1 lo
  g1[3] = (int)((ROWS >> 16) | ((unsigned)TT << 16));      // td1 hi | tile_dim0 = 64
  g1[4] = (int)(ROWS & 0xffffu);                    // tile_dim1 = 32, tile_dim2 = 0
  g1[5] = (int)T_LEN;                               // tensor_dim0_stride lo32 = 4096
  g1[6] = 0;                                        // stride0 hi16 | stride1 lo16
  g1[7] = 0;                                        // stride1 hi32
  i32x4 z4 = {0, 0, 0, 0};                          // groups 2/3 unused (2D)
  i32x8 z8 = {0, 0, 0, 0, 0, 0, 0, 0};              // trailing group unused
  __builtin_amdgcn_tensor_load_to_lds(g0, g1, z4, z4, z8, 0);
}
#endif

// Fallback cooperative tile load (coalesced 128B rows), also host-parse path.
__device__ __forceinline__ void manual_load_tile(float* lds, unsigned buf_dw,
                                                 const float* gptr, int lane) {
  for (int r = 0; r < ROWS; ++r) {
    const float* grow = gptr + (__SIZE_TYPE__)r * T_LEN;
    lds[buf_dw + r * ROWDW + lane]      = grow[lane];
    lds[buf_dw + r * ROWDW + 32 + lane] = grow[32 + lane];
  }
}

// ---------------------------------------------------------------------------
// One wave32 per block; lane L owns sequence s = blockIdx*32 + L = b*C + c.
// Double-buffered TDM loads overlap the serial recurrence of the current tile.
// ---------------------------------------------------------------------------
__global__ __launch_bounds__(32)
void depth_lstm_kernel(const float* __restrict__ x,
                       const float* __restrict__ Wih,
                       const float* __restrict__ Whh,
                       float* __restrict__ out) {
  extern __shared__ float lds[];
  const int lane = threadIdx.x;
  const int s0   = blockIdx.x * ROWS;
  const int s    = s0 + lane;
  const int ch   = s & (C_CH - 1);

  // Per-channel scalar weights; gates [i, f, g, o]. Pre-halve i/f/o weights so
  // sigma(g) = 0.5*tanh(g_half) + 0.5 (one tanh + one fma per sigmoid).
  const float wi0 = Wih[ch * 4 + 0] * 0.5f;
  const float wi1 = Wih[ch * 4 + 1] * 0.5f;
  const float wi2 = Wih[ch * 4 + 2];
  const float wi3 = Wih[ch * 4 + 3] * 0.5f;
  const float wh0 = Whh[ch * 4 + 0] * 0.5f;
  const float wh1 = Whh[ch * 4 + 1] * 0.5f;
  const float wh2 = Whh[ch * 4 + 2];
  const float wh3 = Whh[ch * 4 + 3] * 0.5f;

  const float* xtile_base = x + (__SIZE_TYPE__)s0 * T_LEN;
  float* outp = out + (__SIZE_TYPE__)s * T_LEN;

#if HAVE_TDM
  tdm_load_tile(0u, xtile_base);                    // prologue: tile 0 -> buf 0
#endif

  float h = 0.0f, cs = 0.0f;
  for (int k = 0; k < NTILE; ++k) {
#if HAVE_TDM
    if (k + 1 < NTILE) {
      tdm_load_tile((unsigned)(((k + 1) & 1) * BUFBYTES),
                    xtile_base + (__SIZE_TYPE__)(k + 1) * TT);
      __builtin_amdgcn_s_wait_tensorcnt(1);         // tile k landed (TDM in-order)
    } else {
      __builtin_amdgcn_s_wait_tensorcnt(0);
    }
    const float* xs = lds + (unsigned)((k & 1) * BUFDW) + lane * ROWDW;
#else
    manual_load_tile(lds, 0u, xtile_base + (__SIZE_TYPE__)k * TT, lane);
    const float* xs = lds + lane * ROWDW;
#endif

    for (int t = 0; t < TT; t += 4) {
      f32x4 hv;
#pragma unroll
      for (int j = 0; j < 4; ++j) {
        const float xv = xs[t + j];
        const float gi = fmaf(h, wh0, xv * wi0);    // half-scaled pre-activation
        const float gf = fmaf(h, wh1, xv * wi1);
        const float gg = fmaf(h, wh2, xv * wi2);
        const float go = fmaf(h, wh3, xv * wi3);
        const float i_ = fmaf(fast_tanh(gi), 0.5f, 0.5f);
        const float f_ = fmaf(fast_tanh(gf), 0.5f, 0.5f);
        const float g_ = fast_tanh(gg);
        const float o_ = fmaf(fast_tanh(go), 0.5f, 0.5f);
        cs = fmaf(f_, cs, i_ * g_);
        h  = o_ * fast_tanh(cs);
        hv[j] = h;
      }
      // Per-lane sequential 16B stores: lines become fully dirty in the 192MB
      // L2 before eviction -> full HBM write efficiency without staging.
      *reinterpret_cast<f32x4*>(outp + k * TT + t) = hv;
    }
  }
}

extern "C" void kernel_launch(void* const* d_in, const int* in_sizes, int n_in,
                              void* d_out, int out_size, void* d_ws, size_t ws_size,
                              hipStream_t stream) {
  const float* x   = (const float*)d_in[0];  // (B, C, T)
  const float* Wih = (const float*)d_in[1];  // (C, 4)
  const float* Whh = (const float*)d_in[2];  // (C, 4)
  float* out = (float*)d_out;                // (B, C, T)

  dim3 grid((B_SZ * C_CH) / ROWS);           // 256 single-wave blocks
  dim3 block(ROWS);                          // 32 threads = 1 wave32
  size_t shmem = 2 * BUFBYTES;               // double-buffered input tiles
  hipLaunchKernelGGL(depth_lstm_kernel, grid, block, shmem, stream,
                     x, Wih, Whh, out);
}